// Fusion_4681514352729
// MI455X (gfx1250) — compile-verified
//
#include <hip/hip_runtime.h>
#include <math.h>

typedef __attribute__((ext_vector_type(2))) float v2f;
typedef __attribute__((ext_vector_type(8))) float v8f;

#define NNODES 1500
#define BSZ    4
#define NTILES ((NNODES + 15) / 16)   // 94
#define NEGC   (-9.0e15f)

static __device__ __forceinline__ float lrelu(float x) { return x > 0.f ? x : 0.5f * x; }
static __device__ __forceinline__ float eluf (float x) { return x > 0.f ? x : expm1f(x); }

// ---------------------------------------------------------------------------
// Layer-1 projection: x [B,N,1] -> per-head h [4][B*N][8], s, d  (h = x*W row)
// ---------------------------------------------------------------------------
struct L1P { const float* W[4]; const float* a[4]; };

__global__ void prep_l1(const float* __restrict__ x, L1P p,
                        float* __restrict__ H, float* __restrict__ S,
                        float* __restrict__ D)
{
    int t = blockIdx.x * blockDim.x + threadIdx.x;
    if (t >= BSZ * NNODES) return;
    float xv = x[t];                       // mean over last dim of size 1
    for (int hh = 0; hh < 4; ++hh) {
        const float* W = p.W[hh];
        const float* a = p.a[hh];
        float ws = 0.f, wd = 0.f;
        float* Hb = H + ((size_t)hh * BSZ * NNODES + t) * 8;
        #pragma unroll
        for (int f = 0; f < 8; ++f) {
            float w = W[f];
            Hb[f] = xv * w;
            ws += w * a[f];
            wd += w * a[8 + f];
        }
        S[(size_t)hh * BSZ * NNODES + t] = xv * ws;
        D[(size_t)hh * BSZ * NNODES + t] = xv * wd;
    }
}

// ---------------------------------------------------------------------------
// Layer-2 projection: h1 [B*N,32] -> per-head h2 [3][B*N][16], s, d
// ---------------------------------------------------------------------------
struct L2P { const float* W[3]; const float* a[3]; };

__global__ void prep_l2(const float* __restrict__ h1, L2P p,
                        float* __restrict__ H, float* __restrict__ S,
                        float* __restrict__ D)
{
    int t = blockIdx.x * blockDim.x + threadIdx.x;
    if (t >= 3 * BSZ * NNODES) return;
    int bn = t % (BSZ * NNODES);
    int hh = t / (BSZ * NNODES);
    const float* row = h1 + (size_t)bn * 32;
    const float* W = p.W[hh];
    const float* a = p.a[hh];
    float hv[16];
    #pragma unroll
    for (int f = 0; f < 16; ++f) {
        float acc = 0.f;
        #pragma unroll
        for (int k = 0; k < 32; ++k) acc += row[k] * W[k * 16 + f];
        hv[f] = acc;
    }
    float s = 0.f, d = 0.f;
    float* Hb = H + (size_t)t * 16;
    #pragma unroll
    for (int f = 0; f < 16; ++f) {
        Hb[f] = hv[f];
        s += hv[f] * a[f];
        d += hv[f] * a[16 + f];
    }
    S[t] = s;
    D[t] = d;
}

// ---------------------------------------------------------------------------
// Pass 1: row-wise softmax stats (max, sum) of masked leakyrelu(s_i + d_j).
// One wave (32 lanes) per (head,b,row).
// ---------------------------------------------------------------------------
__global__ void attn_stats(const float* __restrict__ S, const float* __restrict__ D,
                           const int* __restrict__ adj,
                           float* __restrict__ M, float* __restrict__ L, int nrows)
{
    int wave = (int)((blockIdx.x * blockDim.x + threadIdx.x) >> 5);
    int lane = threadIdx.x & 31;
    if (wave >= nrows) return;
    int i  = wave % NNODES;
    int hb = wave / NNODES;
    const float* Db = D + (size_t)hb * NNODES;
    const int* arow = adj + (size_t)i * NNODES;
    float s = S[(size_t)hb * NNODES + i];
    float m = -INFINITY, l = 0.f;
    for (int j = lane; j < NNODES; j += 32) {
        float e  = (arow[j] > 0) ? lrelu(s + Db[j]) : NEGC;
        float mn = fmaxf(m, e);
        l = l * __expf(m - mn) + __expf(e - mn);
        m = mn;
    }
    #pragma unroll
    for (int k = 16; k > 0; k >>= 1) {
        float m2 = __shfl_xor(m, k, 32);
        float l2 = __shfl_xor(l, k, 32);
        float mn = fmaxf(m, m2);
        l = l * __expf(m - mn) + l2 * __expf(m2 - mn);
        m = mn;
    }
    if (lane == 0) { M[wave] = m; L[wave] = l; }
}

// ---------------------------------------------------------------------------
// Pass 2: out[i,f] = ELU( sum_j P_ij * h[j,f] ) via V_WMMA_F32_16X16X4_F32.
// One wave per 16-row tile. A-frag (16x4 f32): lane L holds M=L%16,
// K=2*(L/16)+{0,1}. B-frag (4x16 f32): lane L holds N=L%16, same two K.
// C layout: VGPR r -> rows r (lanes 0-15) / r+8 (lanes 16-31), col = lane%16.
// K-loop is fully branchless so EXEC stays all-ones for WMMA.
// ---------------------------------------------------------------------------
__global__ void __launch_bounds__(32)
attn_pv_wmma(const float* __restrict__ H, const float* __restrict__ Sv,
             const float* __restrict__ Dv, const float* __restrict__ Mv,
             const float* __restrict__ Lv, const int* __restrict__ adj,
             float* __restrict__ OUT, int F, int C)
{
    int lane = threadIdx.x;
    int tile = blockIdx.x % NTILES;
    int hb   = blockIdx.x / NTILES;
    int hh   = hb / BSZ;
    int b    = hb % BSZ;
    int i0   = tile * 16;

    const float* Hb = H  + (size_t)hb * NNODES * F;
    const float* Db = Dv + (size_t)hb * NNODES;

    int   rowA   = i0 + (lane & 15);
    int   rowAc  = rowA < NNODES ? rowA : NNODES - 1;
    float pValid = rowA < NNODES ? 1.f : 0.f;
    float sA   = Sv[(size_t)hb * NNODES + rowAc];
    float mA   = Mv[(size_t)hb * NNODES + rowAc];
    float invl = 1.f / Lv[(size_t)hb * NNODES + rowAc];

    int   col      = lane & 15;
    int   colc     = col < F ? col : 0;
    float colValid = col < F ? 1.f : 0.f;
    int   kOff     = (lane >> 4) << 1;      // 0 or 2
    const int* arow = adj + (size_t)rowAc * NNODES;

    v8f c = {0.f, 0.f, 0.f, 0.f, 0.f, 0.f, 0.f, 0.f};

    #pragma unroll 2
    for (int j0 = 0; j0 < NNODES; j0 += 4) {    // NNODES % 4 == 0
        int jA = j0 + kOff;
        int2   aa = *(const int2*)  (arow + jA);
        float2 dd = *(const float2*)(Db   + jA);
        float e0 = (aa.x > 0) ? lrelu(sA + dd.x) : NEGC;
        float e1 = (aa.y > 0) ? lrelu(sA + dd.y) : NEGC;
        v2f afrag;
        afrag.x = __expf(e0 - mA) * invl * pValid;
        afrag.y = __expf(e1 - mA) * invl * pValid;
        v2f bfrag;
        bfrag.x = Hb[(size_t)jA * F + colc] * colValid;
        bfrag.y = Hb[(size_t)(jA + 1) * F + colc] * colValid;
        c = __builtin_amdgcn_wmma_f32_16x16x4_f32(false, afrag, false, bfrag,
                                                  (short)0, c, false, false);
    }

    int rGroup = (lane >> 4) << 3;              // 0 or 8
    #pragma unroll
    for (int r = 0; r < 8; ++r) {
        int row = i0 + r + rGroup;
        if (row < NNODES && col < F)
            OUT[((size_t)b * NNODES + row) * C + hh * F + col] = eluf(c[r]);
    }
}

// ---------------------------------------------------------------------------
// Pool: x = h[b,n,:] . w + bias  -> written straight into Z at column offset.
// ---------------------------------------------------------------------------
__global__ void pool_lin(const float* __restrict__ h, const float* __restrict__ w,
                         const float* __restrict__ bias, float* __restrict__ Z,
                         int Cdim, int zoff)
{
    int t = blockIdx.x * blockDim.x + threadIdx.x;
    if (t >= BSZ * NNODES) return;
    int b = t / NNODES, n = t % NNODES;
    const float* row = h + (size_t)t * Cdim;
    float acc = bias[0];
    for (int k = 0; k < Cdim; ++k) acc += row[k] * w[k];
    Z[(size_t)b * 4500 + zoff + n] = acc;
}

__global__ void copy_x0(const float* __restrict__ omic, float* __restrict__ Z)
{
    int t = blockIdx.x * blockDim.x + threadIdx.x;
    if (t >= BSZ * NNODES) return;
    int b = t / NNODES, n = t % NNODES;
    Z[(size_t)b * 4500 + n] = omic[t];
}

// ---------------------------------------------------------------------------
// Dense layer, one thread per (b, out).  W reads coalesced across threads.
// act: 0 = none, 1 = ELU, 2 = ReLU
// ---------------------------------------------------------------------------
__global__ void linear_act(const float* __restrict__ X, const float* __restrict__ W,
                           const float* __restrict__ bias, float* __restrict__ Y,
                           int IN, int OUT, int act)
{
    int t = blockIdx.x * blockDim.x + threadIdx.x;
    if (t >= BSZ * OUT) return;
    int b = t / OUT, o = t % OUT;
    const float* x = X + (size_t)b * IN;
    float acc = bias[o];
    for (int k = 0; k < IN; ++k) acc += x[k] * W[(size_t)k * OUT + o];
    if (act == 1)      acc = eluf(acc);
    else if (act == 2) acc = fmaxf(acc, 0.f);
    Y[t] = acc;
}

// ---------------------------------------------------------------------------
// Final fusion: scaled[v] = f_v * (f_v . cw_v + cb_v); out = concat @ mm_w + mm_b
// ---------------------------------------------------------------------------
struct FuseP { const float* cw[3]; const float* cb[3]; const float* mw; const float* mb; };

__global__ void fuse_final(const float* __restrict__ FE /*[3][B][64]*/, FuseP p,
                           float* __restrict__ out)
{
    int t = blockIdx.x * blockDim.x + threadIdx.x;
    if (t >= BSZ * 4) return;
    int b = t / 4, cI = t % 4;
    float acc = p.mb[cI];
    for (int v = 0; v < 3; ++v) {
        const float* f = FE + ((size_t)v * BSZ + b) * 64;
        float conf = p.cb[v][0];
        for (int k = 0; k < 64; ++k) conf += f[k] * p.cw[v][k];
        for (int k = 0; k < 64; ++k) acc += f[k] * conf * p.mw[(size_t)(v * 64 + k) * 4 + cI];
    }
    out[t] = acc;
}

// ---------------------------------------------------------------------------
// Host launch
// ---------------------------------------------------------------------------
extern "C" void kernel_launch(void* const* d_in, const int* in_sizes, int n_in,
                              void* d_out, int out_size, void* d_ws, size_t ws_size,
                              hipStream_t stream)
{
    (void)in_sizes; (void)n_in; (void)out_size; (void)ws_size;
    auto F32 = [&](int i) { return (const float*)d_in[i]; };

    // Workspace carve-up (float offsets, all even -> float2/int2 aligned)
    float* W = (float*)d_ws;
    size_t o = 0;
    float* FEATS = W + o; o += 3 * BSZ * 64;
    float* H1    = W + o; o += (size_t)4 * BSZ * NNODES * 8;
    float* S1    = W + o; o += (size_t)4 * BSZ * NNODES;
    float* D1    = W + o; o += (size_t)4 * BSZ * NNODES;
    float* M1    = W + o; o += (size_t)4 * BSZ * NNODES;
    float* L1v   = W + o; o += (size_t)4 * BSZ * NNODES;
    float* ATT1  = W + o; o += (size_t)BSZ * NNODES * 32;
    float* H2    = W + o; o += (size_t)3 * BSZ * NNODES * 16;
    float* S2    = W + o; o += (size_t)3 * BSZ * NNODES;
    float* D2    = W + o; o += (size_t)3 * BSZ * NNODES;
    float* M2    = W + o; o += (size_t)3 * BSZ * NNODES;
    float* L2v   = W + o; o += (size_t)3 * BSZ * NNODES;
    float* ATT2  = W + o; o += (size_t)BSZ * NNODES * 48;
    float* Z     = W + o; o += (size_t)BSZ * 4500;
    float* Z1    = W + o; o += (size_t)BSZ * 600;
    float* Z2    = W + o; o += (size_t)BSZ * 256;

    const int BN = BSZ * NNODES;          // 6000
    for (int g = 0; g < 3; ++g) {
        const float* omic = F32(g);
        const int*   adj  = (const int*)d_in[3 + g];
        const int G = 14 + 28 * g;        // gat param base (pytree order)

        L1P p1;
        for (int h = 0; h < 4; ++h) { p1.W[h] = F32(G + 10 + 2 * h); p1.a[h] = F32(G + 11 + 2 * h); }
        prep_l1<<<(BN + 255) / 256, 256, 0, stream>>>(omic, p1, H1, S1, D1);

        attn_stats<<<(4 * BN * 32 + 255) / 256, 256, 0, stream>>>(S1, D1, adj, M1, L1v, 4 * BN);
        attn_pv_wmma<<<4 * BSZ * NTILES, 32, 0, stream>>>(H1, S1, D1, M1, L1v, adj, ATT1, 8, 32);

        L2P p2;
        for (int h = 0; h < 3; ++h) { p2.W[h] = F32(G + 18 + 2 * h); p2.a[h] = F32(G + 19 + 2 * h); }
        prep_l2<<<(3 * BN + 255) / 256, 256, 0, stream>>>(ATT1, p2, H2, S2, D2);

        attn_stats<<<(3 * BN * 32 + 255) / 256, 256, 0, stream>>>(S2, D2, adj, M2, L2v, 3 * BN);
        attn_pv_wmma<<<3 * BSZ * NTILES, 32, 0, stream>>>(H2, S2, D2, M2, L2v, adj, ATT2, 16, 48);

        copy_x0 <<<(BN + 255) / 256, 256, 0, stream>>>(omic, Z);
        pool_lin<<<(BN + 255) / 256, 256, 0, stream>>>(ATT1, F32(G + 24), F32(G + 25), Z, 32, 1500);
        pool_lin<<<(BN + 255) / 256, 256, 0, stream>>>(ATT2, F32(G + 26), F32(G + 27), Z, 48, 3000);

        linear_act<<<(BSZ * 600 + 255) / 256, 256, 0, stream>>>(Z,  F32(G + 0), F32(G + 1), Z1, 4500, 600, 1);
        linear_act<<<(BSZ * 256 + 255) / 256, 256, 0, stream>>>(Z1, F32(G + 2), F32(G + 3), Z2, 600, 256, 1);
        // elu(fc3) then relu == relu(fc3); fc4/fc5 are dead for MMlogit.
        linear_act<<<1, 256, 0, stream>>>(Z2, F32(G + 4), F32(G + 5), FEATS + (size_t)g * BSZ * 64, 256, 64, 2);
    }

    FuseP fp;
    for (int v = 0; v < 3; ++v) { fp.cb[v] = F32(6 + v); fp.cw[v] = F32(9 + v); }
    fp.mb = F32(12); fp.mw = F32(13);
    fuse_final<<<1, 64, 0, stream>>>(FEATS, fp, (float*)d_out);
}